// WindowOverlapProcessor_23708219474114
// MI455X (gfx1250) — compile-verified
//
#include <hip/hip_runtime.h>
#include <hip/hip_bf16.h>
#include <stdint.h>

// Problem constants (fixed by setup_inputs in the reference)
#define HH     512
#define WW     512
#define BATCH  8
#define NWIN   3969        // 63*63 windows per batch
#define WROW   63          // windows per row
#define WINB   3072        // bytes per window: 16*16*3*4

// Tile: 8 rows x 64 cols of output per block, 512 threads (16 waves of 32).
// Covering window slots: 2 (h) x 9 (w).  Per slot we async-load only the 8
// needed rows; edge slots (j==0 / j==8) load only the needed half-rows.
// Every window byte is loaded into LDS exactly once across the whole grid:
// total DRAM reads == 97.5 MB (input size) + 25 MB store -> ~5.3 us roofline.
// LDS window cache: 2 planes x 8 rows x 128 packed cols x 3 ch = 6144 f32.

__global__ __launch_bounds__(512) void winoverlap_kernel(
    const float* __restrict__ windows, float* __restrict__ out)
{
    __shared__ __align__(16) float winCache[2 * 8 * 384];  // 24576 B
    __shared__ float gRaw[16];

    const int tid = threadIdx.x;
    const int w0  = blockIdx.x * 64;
    const int h0  = blockIdx.y * 8;
    const int b   = blockIdx.z;

    const unsigned cacheBase = (unsigned)(uintptr_t)(void*)winCache;

    // -------- async bulk copy: global windows -> LDS (16B per lane chunk) ----
    // Per i-plane: j=0 -> 8 rows * 6 chunks, j=1..7 -> 8*12 each, j=8 -> 8*6.
    // Total 2*768 = 1536 chunks == 512 threads * 3.  Issued first so the DMA
    // runs while we build the Gaussian table below.
#pragma unroll
    for (int k = 0; k < 3; ++k) {
        const int unit = tid + (k << 9);          // 0..1535
        const int ip   = (unit >= 768) ? 1 : 0;   // h-slot plane
        const int u    = unit - ip * 768;
        int j, row, rem;
        if (u < 48) {                       // j = 0 (right half rows)
            j = 0; row = u / 6; rem = u - row * 6;
        } else if (u < 720) {               // j = 1..7 (full rows)
            const int t = u - 48;
            j = 1 + t / 96;
            const int t3 = t - (j - 1) * 96;
            row = t3 / 12; rem = t3 - row * 12;
        } else {                            // j = 8 (left half rows)
            j = 8; const int t = u - 720; row = t / 6; rem = t - row * 6;
        }
        const int hs = h0 - 8 + (ip << 3);
        const int ws = w0 - 8 + (j << 3);
        if (hs >= 0 && hs <= 496 && ws >= 0 && ws <= 496) {
            const int win  = (hs >> 3) * WROW + (ws >> 3);
            const int dh0  = h0 - hs;               // 8 (ip=0) or 0 (ip=1)
            const int dwS  = (j == 0) ? 8 : 0;      // half-row start for j==0
            const unsigned gOff = (unsigned)((b * NWIN + win) * WINB
                                   + (dh0 + row) * 192 + dwS * 12 + rem * 16);
            const int colOff = (j == 0) ? 0 : (8 + (j - 1) * 16);
            const unsigned ldsOff = cacheBase
                + (unsigned)((((ip << 3) + row) * 1536) + colOff * 12 + rem * 16);
            asm volatile("global_load_async_to_lds_b128 %0, %1, %2"
                         :
                         : "v"(ldsOff), "v"(gOff), "s"(windows)
                         : "memory");
        }
    }

    // Gaussian (unnormalized) table, sigma = 16/4 = 4 -> exp(-x^2/32),
    // computed while the async copy is in flight.
    if (tid < 16) {
        const float x = (float)tid - 7.5f;
        gRaw[tid] = expf(-(x * x) * (1.0f / 32.0f));
    }

    asm volatile("s_wait_asynccnt 0x0" ::: "memory");
    __syncthreads();

    // -------- gather + blend + normalize ------------------------------------
    float S = 0.0f;
#pragma unroll
    for (int t = 0; t < 16; ++t) S += gRaw[t];
    const float invS = 1.0f / S;

    const int hl = tid >> 6;        // 0..7
    const int wl = tid & 63;        // 0..63
    const int h  = h0 + hl;
    const int w  = w0 + wl;
    const int w7 = wl & 7;
    const int gcolA = ((wl >> 3) << 4) + w7;   // packed col of left window val

    const bool vI0 = (h0 >= 8);     // upper window band valid (uniform)
    const bool vI1 = (h0 <= 496);   // lower window band valid (uniform)
    const bool vA  = (w >= 8);      // left window valid
    const bool vB  = (w < 504);     // right window valid

    const float gh0 = gRaw[hl + 8] * invS;   // dh for ip=0 slot
    const float gh1 = gRaw[hl]     * invS;   // dh for ip=1 slot
    const float gwA = gRaw[w7 + 8] * invS;   // dw for left window
    const float gwB = gRaw[w7]     * invS;   // dw for right window

    const float ghS = (vI0 ? gh0 : 0.0f) + (vI1 ? gh1 : 0.0f);
    const float gwS = (vA ? gwA : 0.0f) + (vB ? gwB : 0.0f);
    const float inv = 1.0f / (ghS * gwS + 1e-8f);

    float a0 = 0.0f, a1 = 0.0f, a2 = 0.0f;
#pragma unroll
    for (int ip = 0; ip < 2; ++ip) {
        const bool vi = ip ? vI1 : vI0;
        if (!vi) continue;
        const float gh = ip ? gh1 : gh0;
        const float* pl = &winCache[((ip << 3) + hl) * 384];
        if (vA) {
            const float wt = gh * gwA;
            const int o = gcolA * 3;
            a0 += wt * pl[o]; a1 += wt * pl[o + 1]; a2 += wt * pl[o + 2];
        }
        if (vB) {
            const float wt = gh * gwB;
            const int o = (gcolA + 8) * 3;
            a0 += wt * pl[o]; a1 += wt * pl[o + 1]; a2 += wt * pl[o + 2];
        }
    }

    const size_t pix = (size_t)h * WW + w;
    const size_t cplane = (size_t)HH * WW;
    float* ob = out + (size_t)b * 3 * cplane;
    ob[pix]              = a0 * inv;
    ob[pix + cplane]     = a1 * inv;
    ob[pix + 2 * cplane] = a2 * inv;
}

extern "C" void kernel_launch(void* const* d_in, const int* in_sizes, int n_in,
                              void* d_out, int out_size, void* d_ws, size_t ws_size,
                              hipStream_t stream) {
    (void)in_sizes; (void)n_in; (void)d_ws; (void)ws_size; (void)out_size;
    const float* windows = (const float*)d_in[0];   // [8,3969,16,16,3] f32
    float* out = (float*)d_out;                      // [8,3,512,512] f32
    dim3 grid(WW / 64, HH / 8, BATCH);               // (8, 64, 8)
    winoverlap_kernel<<<grid, 512, 0, stream>>>(windows, out);
}